// LightGCNWithNG_73220602462343
// MI455X (gfx1250) — compile-verified
//
#include <hip/hip_runtime.h>
#include <hip/hip_bf16.h>

#define NUM_USERS 100000
#define NUM_ITEMS 50000
#define NROWS     (NUM_USERS + NUM_ITEMS)   // 150000
#define DIM       64
#define HOPS      3
#define BATCH     4096

typedef float v2f __attribute__((ext_vector_type(2)));
typedef float v8f __attribute__((ext_vector_type(8)));

// ---------------------------------------------------------------------------
// init: acc = xa = concat(user_emb, item_emb); processed as float4 elements
// ---------------------------------------------------------------------------
__global__ __launch_bounds__(256) void lg_init_kernel(
    const float* __restrict__ user_emb, const float* __restrict__ item_emb,
    float* __restrict__ acc, float* __restrict__ xa)
{
    long i = (long)blockIdx.x * 256 + threadIdx.x;      // float4 index
    const long total = (long)NROWS * (DIM / 4);
    if (i >= total) return;
    const long usplit = (long)NUM_USERS * (DIM / 4);
    float4 v = (i < usplit) ? ((const float4*)user_emb)[i]
                            : ((const float4*)item_emb)[i - usplit];
    ((float4*)acc)[i] = v;
    ((float4*)xa)[i]  = v;
}

// ---------------------------------------------------------------------------
// SpMM: y[row[e]] += val[e] * x[col[e]]   (16 lanes per edge, float4 per lane)
// Coalesced 256B gather per edge (global_load_b128), f32 atomic scatter.
// ---------------------------------------------------------------------------
__global__ __launch_bounds__(256) void lg_spmm_kernel(
    const int* __restrict__ erow, const int* __restrict__ ecol,
    const float* __restrict__ eval,
    const float* __restrict__ x, float* __restrict__ y, int nnz)
{
    long gid = (long)blockIdx.x * 256 + threadIdx.x;
    long e = gid >> 4;
    if (e >= nnz) return;
    int t = (int)(gid & 15);

    int   c = ecol[e];
    int   r = erow[e];
    float v = eval[e];

    float4 d = ((const float4*)(x + (size_t)c * DIM))[t];
    float* dst = y + (size_t)r * DIM + t * 4;
    atomicAdd(dst + 0, v * d.x);
    atomicAdd(dst + 1, v * d.y);
    atomicAdd(dst + 2, v * d.z);
    atomicAdd(dst + 3, v * d.w);
}

// ---------------------------------------------------------------------------
// acc += x (dense, float4)
// ---------------------------------------------------------------------------
__global__ __launch_bounds__(256) void lg_acc_kernel(
    float* __restrict__ acc, const float* __restrict__ x)
{
    long i = (long)blockIdx.x * 256 + threadIdx.x;
    const long total = (long)NROWS * (DIM / 4);
    if (i >= total) return;
    float4 a = ((const float4*)acc)[i];
    float4 b = ((const float4*)x)[i];
    a.x += b.x; a.y += b.y; a.z += b.z; a.w += b.w;
    ((float4*)acc)[i] = a;
}

// ---------------------------------------------------------------------------
// Scoring via V_WMMA_F32_16X16X4_F32.
// One wave handles 16 batch entries: A = U tile (16x64), B = I^T / Neg^T.
// Per K-chunk of 4: lane l (g=l>>4, m=l&15) supplies
//   A vgpr v : U[m][kb + 2g + v]      (ISA 16x4 f32 A layout: K = 2g+v)
//   B vgpr v : I[n][kb + 2g + v], n=m (B rows striped across lanes, same K map)
// D diag: element M lives in vgpr (M&7) of lane M (g=0,M<8) or lane M+16 (g=1).
// acc holds 4*light_out for both operands -> fold 1/16 into the score diff.
// ---------------------------------------------------------------------------
__global__ __launch_bounds__(256) void lg_score_kernel(
    const float* __restrict__ acc,
    const int* __restrict__ users, const int* __restrict__ items,
    const int* __restrict__ negs, float* __restrict__ partials)
{
    __shared__ float s_sum;
    if (threadIdx.x == 0) s_sum = 0.0f;
    __syncthreads();

    const int lane  = threadIdx.x & 31;
    const int wave  = threadIdx.x >> 5;
    const int group = blockIdx.x * 8 + wave;   // 16 batch entries per wave
    const int g = lane >> 4;
    const int m = lane & 15;
    const int j = group * 16 + m;

    const long ru = (long)users[j];
    const long ri = (long)NUM_USERS + items[j];
    const long rn = (long)NUM_USERS + negs[j];

    const float* pu = acc + ru * DIM + 2 * g;
    const float* pi = acc + ri * DIM + 2 * g;
    const float* pn = acc + rn * DIM + 2 * g;

    v8f cpos = {};
    v8f cneg = {};
#pragma unroll
    for (int kc = 0; kc < 16; ++kc) {          // 16 chunks of K=4 cover DIM=64
        v2f a  = *(const v2f*)(pu + 4 * kc);
        v2f bi = *(const v2f*)(pi + 4 * kc);
        v2f bn = *(const v2f*)(pn + 4 * kc);
        cpos = __builtin_amdgcn_wmma_f32_16x16x4_f32(
                   false, a, false, bi, (short)0, cpos, false, false);
        cneg = __builtin_amdgcn_wmma_f32_16x16x4_f32(
                   false, a, false, bn, (short)0, cneg, false, false);
    }

    // Diagonal extraction: lanes 0..7 (g=0) own M=0..7, lanes 24..31 own M=8..15
    bool valid = ((g == 0) && (m < 8)) || ((g == 1) && (m >= 8));
    int  r = m & 7;
    float pos = 0.0f, neg = 0.0f;
#pragma unroll
    for (int rr = 0; rr < 8; ++rr) {
        if (rr == r) { pos = cpos[rr]; neg = cneg[rr]; }
    }

    if (valid) {
        float xdiff = (neg - pos) * (1.0f / 16.0f);  // (1/4)^2 normalization
        // stable softplus
        float sp = fmaxf(xdiff, 0.0f) + log1pf(__expf(-fabsf(xdiff)));
        atomicAdd(&s_sum, sp);
    }
    __syncthreads();
    if (threadIdx.x == 0) partials[blockIdx.x] = s_sum;
}

// ---------------------------------------------------------------------------
// Final reduce: 32 partials -> mean over BATCH
// ---------------------------------------------------------------------------
__global__ void lg_reduce_kernel(const float* __restrict__ partials,
                                 float* __restrict__ out)
{
    float v = partials[threadIdx.x];   // launched with exactly 32 threads
#pragma unroll
    for (int off = 16; off > 0; off >>= 1)
        v += __shfl_down(v, off, 32);
    if (threadIdx.x == 0) out[0] = v * (1.0f / (float)BATCH);
}

// ---------------------------------------------------------------------------
extern "C" void kernel_launch(void* const* d_in, const int* in_sizes, int n_in,
                              void* d_out, int out_size, void* d_ws, size_t ws_size,
                              hipStream_t stream)
{
    const int*   edge_row  = (const int*)d_in[0];
    const int*   edge_col  = (const int*)d_in[1];
    const float* edge_val  = (const float*)d_in[2];
    const float* user_emb  = (const float*)d_in[3];
    const float* item_emb  = (const float*)d_in[4];
    const int*   users     = (const int*)d_in[5];
    const int*   items     = (const int*)d_in[6];
    const int*   negatives = (const int*)d_in[7];
    const int    nnz       = in_sizes[0];

    const size_t tableElems = (size_t)NROWS * DIM;
    float* acc      = (float*)d_ws;
    float* xa       = acc + tableElems;
    float* xb       = xa + tableElems;
    float* partials = xb + tableElems;

    const long total4 = (long)NROWS * (DIM / 4);
    const int  gridDense = (int)((total4 + 255) / 256);

    lg_init_kernel<<<gridDense, 256, 0, stream>>>(user_emb, item_emb, acc, xa);

    float* cur = xa;
    float* nxt = xb;
    for (int h = 0; h < HOPS; ++h) {
        hipMemsetAsync(nxt, 0, tableElems * sizeof(float), stream);
        long nthreads = (long)nnz * 16;
        int  gridSpmm = (int)((nthreads + 255) / 256);
        lg_spmm_kernel<<<gridSpmm, 256, 0, stream>>>(edge_row, edge_col, edge_val,
                                                     cur, nxt, nnz);
        lg_acc_kernel<<<gridDense, 256, 0, stream>>>(acc, nxt);
        float* t = cur; cur = nxt; nxt = t;
    }

    // 8 waves/block * 16 entries/wave = 128 entries per block -> 32 blocks
    lg_score_kernel<<<BATCH / 128, 256, 0, stream>>>(acc, users, items,
                                                     negatives, partials);
    lg_reduce_kernel<<<1, 32, 0, stream>>>(partials, (float*)d_out);
}